// GATEncoder_10282151706779
// MI455X (gfx1250) — compile-verified
//
#include <hip/hip_runtime.h>
#include <math.h>

#define N_NODES 50000
#define E_EDGES 1600000
#define DIN     128
#define HC      128      // H*C
#define NHEAD   4
#define CDIM    32
#define NEG_SLOPE 0.2f
#define LN_EPS  1e-5f

typedef float v2f __attribute__((ext_vector_type(2)));
typedef float v8f __attribute__((ext_vector_type(8)));

// ---------------------------------------------------------------------------
// Kernel 1: init accumulators: out_acc = 0, m = -inf, s = 0
// ---------------------------------------------------------------------------
__global__ void init_bufs(float* __restrict__ out_acc, float* __restrict__ m,
                          float* __restrict__ s) {
    int i = blockIdx.x * blockDim.x + threadIdx.x;
    if (i < N_NODES * HC) out_acc[i] = 0.0f;
    if (i < N_NODES * NHEAD) { m[i] = -INFINITY; s[i] = 0.0f; }
}

// ---------------------------------------------------------------------------
// Kernel 2: dual GEMM h_src = x@Wsrc, h_dst = x@Wdst via V_WMMA_F32_16X16X4_F32
// grid.x = N/16 row tiles, 8 waves per block = 8 column tiles of 16.
// A fragment (x tile) is shared between both accumulators.
// ---------------------------------------------------------------------------
__global__ __launch_bounds__(256) void gemm_dual_wmma(
    const float* __restrict__ x, const float* __restrict__ Wsrc,
    const float* __restrict__ Wdst, float* __restrict__ h_src,
    float* __restrict__ h_dst)
{
    const int lane = threadIdx.x & 31;
    const int ct   = threadIdx.x >> 5;   // column tile 0..7
    const int half = lane >> 4;          // lane half (0: lanes 0-15, 1: 16-31)
    const int l16  = lane & 15;
    const int rowA = blockIdx.x * 16 + l16;      // A-matrix row (M = l16)
    const int col  = ct * 16 + l16;              // B/D column

    v8f acc_s = {0.f,0.f,0.f,0.f,0.f,0.f,0.f,0.f};
    v8f acc_d = {0.f,0.f,0.f,0.f,0.f,0.f,0.f,0.f};

    const float* xrow = x + (size_t)rowA * DIN;

    #pragma unroll 4
    for (int k = 0; k < DIN; k += 4) {
        // A 16x4: VGPR0 = K (+0|+2 per half), VGPR1 = K (+1|+3)
        v2f a;
        a.x = xrow[k + 2 * half];
        a.y = xrow[k + 2 * half + 1];
        // B 4x16: same half-split pattern over K, columns across lanes
        const int r0 = (k + 2 * half) * HC + col;
        v2f bs, bd;
        bs.x = Wsrc[r0];      bs.y = Wsrc[r0 + HC];
        bd.x = Wdst[r0];      bd.y = Wdst[r0 + HC];

        acc_s = __builtin_amdgcn_wmma_f32_16x16x4_f32(
            false, a, false, bs, (short)0, acc_s, false, false);
        acc_d = __builtin_amdgcn_wmma_f32_16x16x4_f32(
            false, a, false, bd, (short)0, acc_d, false, false);
    }

    // C/D 16x16: VGPR v -> row M = v + 8*half, column = lane%16
    #pragma unroll
    for (int v = 0; v < 8; ++v) {
        const int rowM = blockIdx.x * 16 + v + 8 * half;
        h_src[(size_t)rowM * HC + col] = acc_s[v];
        h_dst[(size_t)rowM * HC + col] = acc_d[v];
    }
}

// ---------------------------------------------------------------------------
// Kernel 3: a_src[n,h] = dot(h_src[n,h,:], att_src[h,:]) ; same for dst
// ---------------------------------------------------------------------------
__global__ void att_dots(const float* __restrict__ h_src,
                         const float* __restrict__ h_dst,
                         const float* __restrict__ att_src,
                         const float* __restrict__ att_dst,
                         float* __restrict__ a_src, float* __restrict__ a_dst)
{
    int idx = blockIdx.x * blockDim.x + threadIdx.x;   // n*4 + h
    if (idx >= N_NODES * NHEAD) return;
    const int n = idx >> 2, h = idx & 3;
    const float* hs = h_src + (size_t)n * HC + h * CDIM;
    const float* hd = h_dst + (size_t)n * HC + h * CDIM;
    const float* as = att_src + h * CDIM;
    const float* ad = att_dst + h * CDIM;
    float ss = 0.f, sd = 0.f;
    #pragma unroll
    for (int j = 0; j < CDIM; ++j) { ss += hs[j] * as[j]; sd += hd[j] * ad[j]; }
    a_src[idx] = ss; a_dst[idx] = sd;
}

// ---------------------------------------------------------------------------
// float atomic max via int/uint punning (sign-magnitude ordering trick)
// ---------------------------------------------------------------------------
__device__ inline void atomicMaxF(float* addr, float v) {
    if (v >= 0.0f) atomicMax((int*)addr, __float_as_int(v));
    else           atomicMin((unsigned int*)addr, __float_as_uint(v));
}

__device__ inline float leaky(float v) { return v >= 0.f ? v : NEG_SLOPE * v; }

// ---------------------------------------------------------------------------
// Kernel 4: per-edge logits + segment max into m[dst]
// ---------------------------------------------------------------------------
__global__ void edge_logits(const int* __restrict__ src,
                            const int* __restrict__ dst,
                            const float* __restrict__ a_src,
                            const float* __restrict__ a_dst,
                            float* __restrict__ logits, float* __restrict__ m)
{
    int e = blockIdx.x * blockDim.x + threadIdx.x;
    if (e >= E_EDGES) return;
    const int se = src[e], de = dst[e];
    const float4 as = *(const float4*)(a_src + (size_t)se * 4);
    const float4 ad = *(const float4*)(a_dst + (size_t)de * 4);
    float4 l;
    l.x = leaky(as.x + ad.x); l.y = leaky(as.y + ad.y);
    l.z = leaky(as.z + ad.z); l.w = leaky(as.w + ad.w);
    *(float4*)(logits + (size_t)e * 4) = l;
    float* mp = m + (size_t)de * 4;
    atomicMaxF(mp + 0, l.x); atomicMaxF(mp + 1, l.y);
    atomicMaxF(mp + 2, l.z); atomicMaxF(mp + 3, l.w);
}

// ---------------------------------------------------------------------------
// Kernel 5: e = exp(logit - m[dst]); segment sum into s[dst]; store e in place
// ---------------------------------------------------------------------------
__global__ void edge_exp(const int* __restrict__ dst,
                         const float* __restrict__ m,
                         float* __restrict__ logits, float* __restrict__ s)
{
    int e = blockIdx.x * blockDim.x + threadIdx.x;
    if (e >= E_EDGES) return;
    const int de = dst[e];
    float4 l = *(float4*)(logits + (size_t)e * 4);
    const float4 mm = *(const float4*)(m + (size_t)de * 4);
    float4 ev;
    ev.x = __expf(l.x - mm.x); ev.y = __expf(l.y - mm.y);
    ev.z = __expf(l.z - mm.z); ev.w = __expf(l.w - mm.w);
    *(float4*)(logits + (size_t)e * 4) = ev;
    float* sp = s + (size_t)de * 4;
    atomicAdd(sp + 0, ev.x); atomicAdd(sp + 1, ev.y);
    atomicAdd(sp + 2, ev.z); atomicAdd(sp + 3, ev.w);
}

// ---------------------------------------------------------------------------
// Kernel 6: one wave32 per edge; lane -> 4 channels (float4), head = lane/8.
// Gather h_src[src] (128B coalesced), scale by attn, atomic scatter to out[dst].
// Also emits attention[E,H] output.
// ---------------------------------------------------------------------------
__global__ __launch_bounds__(256) void edge_aggregate(
    const int* __restrict__ src, const int* __restrict__ dst,
    const float* __restrict__ h_src, const float* __restrict__ eval,
    const float* __restrict__ ssum, float* __restrict__ out_acc,
    float* __restrict__ out_attn)
{
    const int lane = threadIdx.x & 31;
    const int e = blockIdx.x * 8 + (threadIdx.x >> 5);
    if (e >= E_EDGES) return;
    const int se = src[e], de = dst[e];
    const int head = lane >> 3;
    const float attn = eval[(size_t)e * 4 + head] /
                       (ssum[(size_t)de * 4 + head] + 1e-16f);
    if ((lane & 7) == 0) out_attn[(size_t)e * 4 + head] = attn;
    const float4 hv = *(const float4*)(h_src + (size_t)se * HC + lane * 4);
    float* o = out_acc + (size_t)de * HC + lane * 4;
    atomicAdd(o + 0, hv.x * attn);
    atomicAdd(o + 1, hv.y * attn);
    atomicAdd(o + 2, hv.z * attn);
    atomicAdd(o + 3, hv.w * attn);
}

// ---------------------------------------------------------------------------
// Kernel 7: bias + LayerNorm + PReLU, one wave32 per node, in place on d_out
// ---------------------------------------------------------------------------
__global__ __launch_bounds__(256) void ln_prelu(
    float* __restrict__ out, const float* __restrict__ bias,
    const float* __restrict__ gamma, const float* __restrict__ beta,
    const float* __restrict__ pw)
{
    const int lane = threadIdx.x & 31;
    const int n = blockIdx.x * 8 + (threadIdx.x >> 5);
    if (n >= N_NODES) return;
    float* row = out + (size_t)n * HC + lane * 4;
    float4 v = *(float4*)row;
    const float4 b = *(const float4*)(bias + lane * 4);
    v.x += b.x; v.y += b.y; v.z += b.z; v.w += b.w;
    float sum = v.x + v.y + v.z + v.w;
    float sq  = v.x*v.x + v.y*v.y + v.z*v.z + v.w*v.w;
    #pragma unroll
    for (int off = 16; off > 0; off >>= 1) {
        sum += __shfl_xor(sum, off, 32);
        sq  += __shfl_xor(sq,  off, 32);
    }
    const float mean = sum * (1.0f / HC);
    const float var  = sq * (1.0f / HC) - mean * mean;
    const float inv  = rsqrtf(var + LN_EPS);
    const float4 g  = *(const float4*)(gamma + lane * 4);
    const float4 be = *(const float4*)(beta  + lane * 4);
    const float4 p  = *(const float4*)(pw    + lane * 4);
    float4 r;
    float t;
    t = (v.x - mean) * inv * g.x + be.x; r.x = t >= 0.f ? t : p.x * t;
    t = (v.y - mean) * inv * g.y + be.y; r.y = t >= 0.f ? t : p.y * t;
    t = (v.z - mean) * inv * g.z + be.z; r.z = t >= 0.f ? t : p.z * t;
    t = (v.w - mean) * inv * g.w + be.w; r.w = t >= 0.f ? t : p.w * t;
    *(float4*)row = r;
}

// ---------------------------------------------------------------------------
extern "C" void kernel_launch(void* const* d_in, const int* in_sizes, int n_in,
                              void* d_out, int out_size, void* d_ws, size_t ws_size,
                              hipStream_t stream) {
    const int*   edge_index = (const int*)  d_in[0];   // [2,E] int32
    const float* x          = (const float*)d_in[1];
    // d_in[2] = edge_attr (unused)
    const float* W_src      = (const float*)d_in[3];
    const float* W_dst      = (const float*)d_in[4];
    const float* att_src    = (const float*)d_in[5];
    const float* att_dst    = (const float*)d_in[6];
    const float* bias       = (const float*)d_in[7];
    const float* ln_gamma   = (const float*)d_in[8];
    const float* ln_beta    = (const float*)d_in[9];
    const float* prelu_w    = (const float*)d_in[10];

    const int* src = edge_index;
    const int* dst = edge_index + E_EDGES;

    float* x_after  = (float*)d_out;                    // [N, 128]
    float* out_attn = (float*)d_out + (size_t)N_NODES * HC; // [E, 4]

    // Workspace layout (256B aligned blocks)
    char* ws = (char*)d_ws;
    auto alignup = [](size_t v) { return (v + 255) & ~(size_t)255; };
    size_t off = 0;
    float* h_src = (float*)(ws + off); off += alignup((size_t)N_NODES * HC * 4);
    float* h_dst = (float*)(ws + off); off += alignup((size_t)N_NODES * HC * 4);
    float* a_src = (float*)(ws + off); off += alignup((size_t)N_NODES * NHEAD * 4);
    float* a_dst = (float*)(ws + off); off += alignup((size_t)N_NODES * NHEAD * 4);
    float* mbuf  = (float*)(ws + off); off += alignup((size_t)N_NODES * NHEAD * 4);
    float* sbuf  = (float*)(ws + off); off += alignup((size_t)N_NODES * NHEAD * 4);
    float* logit = (float*)(ws + off); off += alignup((size_t)E_EDGES * NHEAD * 4);

    // 1. init out accumulator / segment max / segment sum
    init_bufs<<<(N_NODES * HC + 255) / 256, 256, 0, stream>>>(x_after, mbuf, sbuf);
    // 2. dual GEMM via f32 WMMA
    gemm_dual_wmma<<<N_NODES / 16, 256, 0, stream>>>(x, W_src, W_dst, h_src, h_dst);
    // 3. attention dot products
    att_dots<<<(N_NODES * NHEAD + 255) / 256, 256, 0, stream>>>(
        h_src, h_dst, att_src, att_dst, a_src, a_dst);
    // 4. edge logits + segment max
    edge_logits<<<(E_EDGES + 255) / 256, 256, 0, stream>>>(
        src, dst, a_src, a_dst, logit, mbuf);
    // 5. exp + segment sum
    edge_exp<<<(E_EDGES + 255) / 256, 256, 0, stream>>>(dst, mbuf, logit, sbuf);
    // 6. scatter-aggregate messages (wave per edge) + attention output
    edge_aggregate<<<E_EDGES / 8, 256, 0, stream>>>(
        src, dst, h_src, logit, sbuf, x_after, out_attn);
    // 7. bias + LayerNorm + PReLU (wave per node), in place
    ln_prelu<<<N_NODES / 8, 256, 0, stream>>>(
        x_after, bias, ln_gamma, ln_beta, prelu_w);
}